// MetadataPredictionFromCtx_29265907155027
// MI455X (gfx1250) — compile-verified
//
#include <hip/hip_runtime.h>
#include <hip/hip_bf16.h>
#include <math.h>

// ---- problem constants (match reference) ----
#define N_CELLS   500000
#define N_SAMPLES 512
#define CTX       256
#define PHENO     128
#define N_DIS     2
#define N_LOBES   5
#define CONS_W    0.1f

typedef __attribute__((ext_vector_type(2))) float v2f;
typedef __attribute__((ext_vector_type(8))) float v8f;

// ---------------------------------------------------------------------------
// Zero a float region
// ---------------------------------------------------------------------------
__global__ void zero_f32(float* __restrict__ p, int n) {
  int i = blockIdx.x * blockDim.x + threadIdx.x;
  if (i < n) p[i] = 0.0f;
}

// ---------------------------------------------------------------------------
// fp32 WMMA GEMM: C[M,N] = act(A[M,K] @ B[K,N] + bias)
// One wave32 per 16x16 output tile, K stepped by 4 via V_WMMA_F32_16X16X4_F32.
// A fragment per ISA layout: lanes 0-15 -> K = k0+0,k0+1 ; lanes 16-31 -> K = k0+2,k0+3
// B / C / D: row per VGPR striped across lanes (lanes 0-15 = N, hi half shifts M/K).
// M, N multiples of 16; K multiple of 4. EXEC stays all-ones per wave.
// ---------------------------------------------------------------------------
__global__ __launch_bounds__(128) void wmma_gemm_f32(
    const float* __restrict__ A, const float* __restrict__ B,
    const float* __restrict__ bias, float* __restrict__ C,
    int M, int N, int K, int do_relu) {
  int wave = (blockIdx.x * blockDim.x + threadIdx.x) >> 5;
  int lane = threadIdx.x & 31;
  int tilesN = N >> 4;
  int tm = wave / tilesN;
  int tn = wave % tilesN;
  if (tm * 16 >= M) return;            // wave-uniform
  int hi = lane >> 4;                  // 0 or 1
  int r  = lane & 15;
  int m0 = tm << 4, n0 = tn << 4;

  v8f acc = {};
  for (int k0 = 0; k0 < K; k0 += 4) {
    int ka = k0 + 2 * hi;
    v2f a, b;
    a.x = A[(size_t)(m0 + r) * K + ka];
    a.y = A[(size_t)(m0 + r) * K + ka + 1];
    b.x = B[(size_t)ka       * N + n0 + r];
    b.y = B[(size_t)(ka + 1) * N + n0 + r];
    acc = __builtin_amdgcn_wmma_f32_16x16x4_f32(
        /*neg_a=*/false, a, /*neg_b=*/false, b,
        /*c_mod=*/(short)0, acc, /*reuse_a=*/false, /*reuse_b=*/false);
  }
#pragma unroll
  for (int v = 0; v < 8; ++v) {
    int m = m0 + v + 8 * hi;
    int n = n0 + r;
    float x = acc[v];
    if (bias) x += bias[n];
    if (do_relu) x = fmaxf(x, 0.0f);
    C[(size_t)m * N + n] = x;
  }
}

// ---------------------------------------------------------------------------
// Per-sample L2 norm of proj rows (one wave per sample)
// ---------------------------------------------------------------------------
__global__ __launch_bounds__(256) void proj_norm_kernel(
    const float* __restrict__ proj, float* __restrict__ proj_norm, int n_samples) {
  int w = (blockIdx.x * blockDim.x + threadIdx.x) >> 5;
  int lane = threadIdx.x & 31;
  if (w >= n_samples) return;
  const float* row = proj + (size_t)w * CTX;
  float s = 0.0f;
  for (int i = lane; i < CTX; i += 32) { float v = row[i]; s += v * v; }
  for (int off = 16; off; off >>= 1) s += __shfl_xor(s, off, 32);
  if (lane == 0) proj_norm[w] = sqrtf(s);
}

// ---------------------------------------------------------------------------
// Fused per-cell streaming pass (the 512 MB roofline pass):
// one wave per cell; coalesced float4 reads of the 1KB z_ctx row.
//  - segment-sum into z_sum[sample][:] + counts[sample]  (L2-resident atomics)
//  - cosine numerator dot(z_ctx, proj[sample]) and ||z_ctx||, wave-reduced
// ---------------------------------------------------------------------------
__global__ __launch_bounds__(256) void cell_pass(
    const float* __restrict__ z_ctx, const int* __restrict__ c2s,
    const float* __restrict__ proj, const float* __restrict__ proj_norm,
    float* __restrict__ z_sum, float* __restrict__ counts,
    float* __restrict__ consist_acc, int n_cells) {
  int cell = (blockIdx.x * blockDim.x + threadIdx.x) >> 5;
  int lane = threadIdx.x & 31;
  if (cell >= n_cells) return;
  int s = c2s[cell];

  const float4* zr = (const float4*)(z_ctx + (size_t)cell * CTX);
  const float4* pr = (const float4*)(proj + (size_t)s * CTX);
  float4 z0 = zr[lane * 2 + 0];
  float4 z1 = zr[lane * 2 + 1];
  float4 p0 = pr[lane * 2 + 0];
  float4 p1 = pr[lane * 2 + 1];

  float dot = z0.x * p0.x + z0.y * p0.y + z0.z * p0.z + z0.w * p0.w
            + z1.x * p1.x + z1.y * p1.y + z1.z * p1.z + z1.w * p1.w;
  float nrm = z0.x * z0.x + z0.y * z0.y + z0.z * z0.z + z0.w * z0.w
            + z1.x * z1.x + z1.y * z1.y + z1.z * z1.z + z1.w * z1.w;

  // segment sum: 8 fp32 atomics per lane into the L2-resident 512KB z_sum
  float* dst = z_sum + (size_t)s * CTX + lane * 8;
  atomicAdd(dst + 0, z0.x); atomicAdd(dst + 1, z0.y);
  atomicAdd(dst + 2, z0.z); atomicAdd(dst + 3, z0.w);
  atomicAdd(dst + 4, z1.x); atomicAdd(dst + 5, z1.y);
  atomicAdd(dst + 6, z1.z); atomicAdd(dst + 7, z1.w);

  for (int off = 16; off; off >>= 1) {
    dot += __shfl_xor(dot, off, 32);
    nrm += __shfl_xor(nrm, off, 32);
  }
  if (lane == 0) {
    float denom = fmaxf(sqrtf(nrm) * proj_norm[s], 1e-8f);
    atomicAdd(consist_acc, dot / denom);
    atomicAdd(&counts[s], 1.0f);
  }
}

// ---------------------------------------------------------------------------
// z_mean = z_sum / max(counts, 1)
// ---------------------------------------------------------------------------
__global__ void finalize_mean(const float* __restrict__ z_sum,
                              const float* __restrict__ counts,
                              float* __restrict__ z_mean, int total) {
  int i = blockIdx.x * blockDim.x + threadIdx.x;
  if (i < total) z_mean[i] = z_sum[i] / fmaxf(counts[i >> 8], 1.0f);
}

// ---------------------------------------------------------------------------
// Small-N second layer of each head: out[m,n] = h[m,:]·W2[:,n] + b2[n]
// ---------------------------------------------------------------------------
__global__ void head_out(const float* __restrict__ h, const float* __restrict__ W2,
                         const float* __restrict__ b2, float* __restrict__ out,
                         int M, int N) {
  int idx = blockIdx.x * blockDim.x + threadIdx.x;
  if (idx >= M * N) return;
  int m = idx / N, n = idx % N;
  float acc = b2[n];
  const float* hr = h + (size_t)m * CTX;
  for (int k = 0; k < CTX; ++k) acc += hr[k] * W2[k * N + n];
  out[idx] = acc;
}

// ---------------------------------------------------------------------------
// Final scalar loss: CE(disease) + CE(lobe) + MSE(age) + 0.1*(1 - mean cos)
// Single block, per-sample contributions, LDS tree reduce.
// ---------------------------------------------------------------------------
__global__ __launch_bounds__(512) void loss_kernel(
    const float* __restrict__ logits_d, const int* __restrict__ lab_d,
    const float* __restrict__ logits_l, const int* __restrict__ lab_l,
    const float* __restrict__ age_pred, const float* __restrict__ age_t,
    const float* __restrict__ consist_acc, float* __restrict__ out,
    int n_samples, float n_cells_f) {
  __shared__ float sh[512];
  int i = threadIdx.x;
  float v = 0.0f;
  if (i < n_samples) {
    // disease CE (2 classes)
    const float* ld = logits_d + i * N_DIS;
    float md = fmaxf(ld[0], ld[1]);
    float lse_d = md + logf(expf(ld[0] - md) + expf(ld[1] - md));
    v += lse_d - ld[lab_d[i]];
    // lobe CE (5 classes)
    const float* ll = logits_l + i * N_LOBES;
    float ml = ll[0];
    for (int c = 1; c < N_LOBES; ++c) ml = fmaxf(ml, ll[c]);
    float se = 0.0f;
    for (int c = 0; c < N_LOBES; ++c) se += expf(ll[c] - ml);
    v += ml + logf(se) - ll[lab_l[i]];
    // age MSE
    float d = age_pred[i] - age_t[i];
    v += d * d;
  }
  sh[i] = v;
  __syncthreads();
  for (int off = 256; off; off >>= 1) {
    if (i < off) sh[i] += sh[i + off];
    __syncthreads();
  }
  if (i == 0) {
    float total = sh[0] / (float)n_samples;
    total += CONS_W * (1.0f - consist_acc[0] / n_cells_f);
    out[0] = total;
  }
}

// ---------------------------------------------------------------------------
// host-side orchestration
// ---------------------------------------------------------------------------
extern "C" void kernel_launch(void* const* d_in, const int* in_sizes, int n_in,
                              void* d_out, int out_size, void* d_ws, size_t ws_size,
                              hipStream_t stream) {
  const float* z_ctx   = (const float*)d_in[0];
  const float* z_pheno = (const float*)d_in[1];
  const int*   c2s     = (const int*)d_in[2];
  const int*   lab_d   = (const int*)d_in[3];
  const int*   lab_l   = (const int*)d_in[4];
  const float* age_t   = (const float*)d_in[5];
  const float* dW1 = (const float*)d_in[6];
  const float* db1 = (const float*)d_in[7];
  const float* dW2 = (const float*)d_in[8];
  const float* db2 = (const float*)d_in[9];
  const float* lW1 = (const float*)d_in[10];
  const float* lb1 = (const float*)d_in[11];
  const float* lW2 = (const float*)d_in[12];
  const float* lb2 = (const float*)d_in[13];
  const float* aW1 = (const float*)d_in[14];
  const float* ab1 = (const float*)d_in[15];
  const float* aW2 = (const float*)d_in[16];
  const float* ab2 = (const float*)d_in[17];
  const float* alignW = (const float*)d_in[18];
  float* out = (float*)d_out;

  // workspace layout (floats)
  float* ws = (float*)d_ws;
  float* proj      = ws;                      // 512*256
  float* proj_nrm  = proj + N_SAMPLES * CTX;  // 512
  float* z_sum     = proj_nrm + N_SAMPLES;    // 512*256
  float* counts    = z_sum + N_SAMPLES * CTX; // 512
  float* z_mean    = counts + N_SAMPLES;      // 512*256
  float* hid_d     = z_mean + N_SAMPLES * CTX;// 512*256
  float* hid_l     = hid_d + N_SAMPLES * CTX; // 512*256
  float* hid_a     = hid_l + N_SAMPLES * CTX; // 512*256
  float* logits_d  = hid_a + N_SAMPLES * CTX; // 512*2
  float* logits_l  = logits_d + N_SAMPLES * N_DIS;   // 512*5
  float* age_pred  = logits_l + N_SAMPLES * N_LOBES; // 512
  float* consist   = age_pred + N_SAMPLES;           // 1

  // 1) zero accumulators (z_sum + counts contiguous, plus consist scalar)
  int nzero = N_SAMPLES * CTX + N_SAMPLES;
  zero_f32<<<(nzero + 255) / 256, 256, 0, stream>>>(z_sum, nzero);
  zero_f32<<<1, 32, 0, stream>>>(consist, 1);

  // 2) proj = z_pheno @ alignW   [512x128]@[128x256] via fp32 WMMA
  {
    int waves = (N_SAMPLES / 16) * (CTX / 16);       // 512
    wmma_gemm_f32<<<(waves * 32) / 128, 128, 0, stream>>>(
        z_pheno, alignW, nullptr, proj, N_SAMPLES, CTX, PHENO, 0);
  }

  // 3) per-sample proj norms
  proj_norm_kernel<<<(N_SAMPLES * 32 + 255) / 256, 256, 0, stream>>>(
      proj, proj_nrm, N_SAMPLES);

  // 4) fused per-cell pass (streaming 512 MB roofline pass)
  {
    int waves = N_CELLS;
    int threads = 256;                                // 8 waves / block
    int blocks = (waves * 32 + threads - 1) / threads;
    cell_pass<<<blocks, threads, 0, stream>>>(
        z_ctx, c2s, proj, proj_nrm, z_sum, counts, consist, N_CELLS);
  }

  // 5) z_mean = z_sum / max(counts,1)
  {
    int total = N_SAMPLES * CTX;
    finalize_mean<<<(total + 255) / 256, 256, 0, stream>>>(z_sum, counts, z_mean, total);
  }

  // 6) head hidden layers: relu(z_mean @ W1 + b1)  [512x256]@[256x256], WMMA
  {
    int waves = (N_SAMPLES / 16) * (CTX / 16);       // 512
    int blocks = (waves * 32) / 128;
    wmma_gemm_f32<<<blocks, 128, 0, stream>>>(z_mean, dW1, db1, hid_d,
                                              N_SAMPLES, CTX, CTX, 1);
    wmma_gemm_f32<<<blocks, 128, 0, stream>>>(z_mean, lW1, lb1, hid_l,
                                              N_SAMPLES, CTX, CTX, 1);
    wmma_gemm_f32<<<blocks, 128, 0, stream>>>(z_mean, aW1, ab1, hid_a,
                                              N_SAMPLES, CTX, CTX, 1);
  }

  // 7) small-N output layers
  head_out<<<(N_SAMPLES * N_DIS + 255) / 256, 256, 0, stream>>>(
      hid_d, dW2, db2, logits_d, N_SAMPLES, N_DIS);
  head_out<<<(N_SAMPLES * N_LOBES + 255) / 256, 256, 0, stream>>>(
      hid_l, lW2, lb2, logits_l, N_SAMPLES, N_LOBES);
  head_out<<<(N_SAMPLES * 1 + 255) / 256, 256, 0, stream>>>(
      hid_a, aW2, ab2, age_pred, N_SAMPLES, 1);

  // 8) final scalar loss
  loss_kernel<<<1, 512, 0, stream>>>(logits_d, lab_d, logits_l, lab_l,
                                     age_pred, age_t, consist, out,
                                     N_SAMPLES, (float)N_CELLS);
}